// FocalLoss_59734405153157
// MI455X (gfx1250) — compile-verified
//
#include <hip/hip_runtime.h>
#include <hip/hip_bf16.h>

// FocalLoss (RetinaNet style) for MI455X / gfx1250, wave32.
// d_in: [0] classifications [B,A,C] f32, [1] regressions [B,A,4] f32,
//       [2] anchors [1,A,4] f32, [3] annotations [B,M,5] f32.
// d_out: 2 floats (mean cls loss, mean reg loss).

typedef float v2f __attribute__((ext_vector_type(2)));
typedef float v8f __attribute__((ext_vector_type(8)));

#define BLOCK 256
#define MAXM  64

// Wave32 all-reduce using V_WMMA_F32_16X16X4_F32.
// A (16x4 f32, 2 VGPRs/lane): v0 = per-lane partial, v1 = 0. B = all-ones.
// D[m][n] = rowsum(A, m) for every n, so each lane's 8 D regs hold 8 rowsums;
// summing them gives half-total in lanes 0-15 and the other half in 16-31.
// One xor-16 shuffle completes the 32-lane sum (result in every lane).
__device__ __forceinline__ float wave_allreduce_wmma(float x) {
    v2f a;    a.x = x;    a.y = 0.0f;
    v2f ones; ones.x = 1.0f; ones.y = 1.0f;
    v8f c = {};
    v8f d = __builtin_amdgcn_wmma_f32_16x16x4_f32(
        /*neg_a=*/false, a, /*neg_b=*/false, ones,
        /*c_mod=*/(short)0, c, /*reuse_a=*/false, /*reuse_b=*/false);
    float s = d[0] + d[1] + d[2] + d[3] + d[4] + d[5] + d[6] + d[7];
    s += __shfl_xor(s, 16, 32);
    return s;
}

__device__ __forceinline__ void block_reduce3_atomic(
    float v0, float v1, float v2, float* g0, float* g1, float* g2) {
    __shared__ float r0[BLOCK / 32], r1[BLOCK / 32], r2[BLOCK / 32];
    // No divergence here: every thread of the block participates (EXEC all-1s
    // is required by WMMA).
    v0 = wave_allreduce_wmma(v0);
    v1 = wave_allreduce_wmma(v1);
    v2 = wave_allreduce_wmma(v2);
    const int lane = threadIdx.x & 31;
    const int w    = threadIdx.x >> 5;
    if (lane == 0) { r0[w] = v0; r1[w] = v1; r2[w] = v2; }
    __syncthreads();
    if (threadIdx.x == 0) {
        float s0 = 0.f, s1 = 0.f, s2 = 0.f;
        const int nw = BLOCK / 32;
        for (int i = 0; i < nw; ++i) { s0 += r0[i]; s1 += r1[i]; s2 += r2[i]; }
        atomicAdd(g0, s0);
        atomicAdd(g1, s1);
        atomicAdd(g2, s2);
    }
}

__global__ void focal_init(float* acc, int n) {
    int i = blockIdx.x * blockDim.x + threadIdx.x;
    if (i < n) acc[i] = 0.0f;
}

__global__ __launch_bounds__(BLOCK) void focal_main(
    const float* __restrict__ cls, const float* __restrict__ reg,
    const float* __restrict__ anc, const float* __restrict__ ann,
    float* __restrict__ acc, int A, int C, int M, int B, int tilesPerImage) {
    const int b     = blockIdx.x / tilesPerImage;
    const int tile  = blockIdx.x - b * tilesPerImage;
    const int aBase = tile * BLOCK;
    const int tid   = threadIdx.x;

    __shared__ float sx1[MAXM], sy1[MAXM], sx2[MAXM], sy2[MAXM], sarea[MAXM];
    __shared__ int   scls[MAXM], svalid[MAXM];
    __shared__ int   sStat[BLOCK];   // class id if pos, -1 neg, -2 ignore

    // Stage annotations for this image into LDS.
    if (tid < M) {
        const float* p = ann + ((size_t)b * M + tid) * 5;
        float x1 = p[0], y1 = p[1], x2 = p[2], y2 = p[3], cv = p[4];
        sx1[tid] = x1; sy1[tid] = y1; sx2[tid] = x2; sy2[tid] = y2;
        sarea[tid]  = (x2 - x1) * (y2 - y1);
        scls[tid]   = (int)cv;
        svalid[tid] = (cv >= 0.0f) ? 1 : 0;
    }
    __syncthreads();

    // ---- Phase 1: per-anchor assignment + regression loss ----
    const int a = aBase + tid;
    float regPart = 0.0f, cntPart = 0.0f;
    int   stat    = -2;
    if (a < A) {
        const float4 ab = *(const float4*)(anc + (size_t)a * 4);
        const float areaA = (ab.z - ab.x) * (ab.w - ab.y);
        float best = -2.0f;
        int   bi   = 0;
        for (int m = 0; m < M; ++m) {
            float v;
            if (svalid[m]) {
                float lx = fmaxf(ab.x, sx1[m]);
                float ly = fmaxf(ab.y, sy1[m]);
                float rx = fminf(ab.z, sx2[m]);
                float ry = fminf(ab.w, sy2[m]);
                float iw = fmaxf(rx - lx, 0.0f);
                float ih = fmaxf(ry - ly, 0.0f);
                float inter = iw * ih;
                v = inter / (areaA + sarea[m] - inter);
            } else {
                v = -1.0f;
            }
            if (v > best) { best = v; bi = m; }   // first-max, like jnp.argmax
        }
        const bool pos = best >= 0.5f;
        const bool neg = best < 0.4f;
        if (pos) {
            stat    = scls[bi];
            cntPart = 1.0f;
            const float aw  = ab.z - ab.x, ah = ab.w - ab.y;
            const float acx = ab.x + 0.5f * aw, acy = ab.y + 0.5f * ah;
            float gw  = sx2[bi] - sx1[bi];
            float gh  = sy2[bi] - sy1[bi];
            const float gcx = sx1[bi] + 0.5f * gw;
            const float gcy = sy1[bi] + 0.5f * gh;
            gw = fmaxf(gw, 1.0f);
            gh = fmaxf(gh, 1.0f);
            const float4 r = *(const float4*)(reg + ((size_t)b * A + a) * 4);
            // /STD with STD = [0.1, 0.1, 0.2, 0.2]
            const float t0 = ((gcx - acx) / aw) * 10.0f;
            const float t1 = ((gcy - acy) / ah) * 10.0f;
            const float t2 = __logf(gw / aw) * 5.0f;
            const float t3 = __logf(gh / ah) * 5.0f;
            const float TH = 1.0f / 9.0f, HB = 0.5f / 9.0f;
            const float d0 = fabsf(t0 - r.x), d1 = fabsf(t1 - r.y);
            const float d2 = fabsf(t2 - r.z), d3 = fabsf(t3 - r.w);
            regPart  = (d0 <= TH) ? 4.5f * d0 * d0 : d0 - HB;
            regPart += (d1 <= TH) ? 4.5f * d1 * d1 : d1 - HB;
            regPart += (d2 <= TH) ? 4.5f * d2 * d2 : d2 - HB;
            regPart += (d3 <= TH) ? 4.5f * d3 * d3 : d3 - HB;
        } else if (neg) {
            stat = -1;
        }
    }
    sStat[tid] = stat;
    __syncthreads();

    // ---- Phase 2: focal classification loss, coalesced stream ----
    const int tileAnch = min(BLOCK, A - aBase);
    const int nElem    = tileAnch * C;
    const float* base  = cls + ((size_t)b * A + aBase) * C;
    float clsPart = 0.0f;

    if ((C & 3) == 0) {
        // float4 groups never straddle an anchor row (C % 4 == 0).
        for (int e = tid * 4; e < nElem; e += BLOCK * 4) {
            const float4 p4 = *(const float4*)(base + e);
            const int al = e / C;
            const int c0 = e - al * C;
            const int s  = sStat[al];
            const float pv[4] = {p4.x, p4.y, p4.z, p4.w};
#pragma unroll
            for (int j = 0; j < 4; ++j) {
                float pc  = fminf(fmaxf(pv[j], 1e-4f), 1.0f - 1e-4f);
                bool  is1 = (s == c0 + j);
                float q   = is1 ? pc : 1.0f - pc;     // single-log formulation
                float af  = is1 ? 0.25f : 0.75f;
                float om  = 1.0f - q;
                float l   = af * om * om * (-__logf(q));
                clsPart += (s == -2) ? 0.0f : l;
            }
        }
    } else {
        for (int e = tid; e < nElem; e += BLOCK) {
            const int al = e / C;
            const int c0 = e - al * C;
            const int s  = sStat[al];
            float pc  = fminf(fmaxf(base[e], 1e-4f), 1.0f - 1e-4f);
            bool  is1 = (s == c0);
            float q   = is1 ? pc : 1.0f - pc;
            float af  = is1 ? 0.25f : 0.75f;
            float om  = 1.0f - q;
            float l   = af * om * om * (-__logf(q));
            clsPart += (s == -2) ? 0.0f : l;
        }
    }

    // ---- Reduce & accumulate per-image (WMMA wave reduction) ----
    block_reduce3_atomic(clsPart, regPart, cntPart,
                         acc + b, acc + B + b, acc + 2 * B + b);
}

__global__ void focal_finalize(const float* __restrict__ acc,
                               float* __restrict__ out, int B) {
    if (blockIdx.x == 0 && threadIdx.x == 0) {
        float cm = 0.0f, rm = 0.0f;
        for (int b = 0; b < B; ++b) {
            float np = acc[2 * B + b];
            cm += acc[b] / fmaxf(np, 1.0f);
            rm += (np > 0.0f) ? acc[B + b] / (np * 4.0f) : 0.0f;
        }
        out[0] = cm / (float)B;
        out[1] = rm / (float)B;
    }
}

extern "C" void kernel_launch(void* const* d_in, const int* in_sizes, int n_in,
                              void* d_out, int out_size, void* d_ws, size_t ws_size,
                              hipStream_t stream) {
    const float* cls = (const float*)d_in[0];
    const float* reg = (const float*)d_in[1];
    const float* anc = (const float*)d_in[2];
    const float* ann = (const float*)d_in[3];

    const int A = in_sizes[2] / 4;              // anchors [1,A,4]
    const int B = in_sizes[1] / (A * 4);        // regressions [B,A,4]
    const int C = in_sizes[0] / (A * B);        // classifications [B,A,C]
    int       M = in_sizes[3] / (B * 5);        // annotations [B,M,5]
    if (M > MAXM) M = MAXM;

    float* acc = (float*)d_ws;                  // [cls_sum[B] | reg_sum[B] | npos[B]]

    focal_init<<<1, 256, 0, stream>>>(acc, 3 * B);

    const int tiles = (A + BLOCK - 1) / BLOCK;  // 384 for A=98304
    focal_main<<<B * tiles, BLOCK, 0, stream>>>(cls, reg, anc, ann, acc,
                                                A, C, M, B, tiles);

    focal_finalize<<<1, 32, 0, stream>>>(acc, (float*)d_out, B);
}